// InverseClassFrequencyLoss_59631325938067
// MI455X (gfx1250) — compile-verified
//
#include <hip/hip_runtime.h>
#include <math.h>

// ---------------------------------------------------------------------------
// InverseClassFrequencyLoss, single-pass fused (per-class sufficient stats).
//
// Roofline: 256 MB streamed once @ 23.3 TB/s => ~11 us floor. ~64M
// transcendentals is far under VALU capacity => memory-bound. No matmul
// structure => WMMA not applicable; the gfx1250 path that matters here is the
// async global->LDS pipeline (ASYNCcnt-tracked, VGPR-free in-flight loads).
// ---------------------------------------------------------------------------

#define AS1 __attribute__((address_space(1)))
#define AS3 __attribute__((address_space(3)))

typedef int v4i __attribute__((ext_vector_type(4)));

#define ICF_BLOCK   256
#define ICF_GRID    2048
#define CHUNK_FLTS  1024   // per array, per block-iteration (256 lanes x 4 floats)

// ---- CDNA5 async global -> LDS copy of 16 bytes (per lane) ----------------
__device__ __forceinline__ void async_ld16(const void* gptr, void* lptr) {
#if __has_builtin(__builtin_amdgcn_global_load_async_to_lds_b128)
  __builtin_amdgcn_global_load_async_to_lds_b128(
      (AS1 v4i*)gptr, (AS3 v4i*)lptr, /*offset=*/0, /*cpol=*/0);
#else
  unsigned lds_off = (unsigned)(unsigned long long)lptr;
  asm volatile("global_load_async_to_lds_b128 %0, %1, off"
               :: "v"(lds_off), "v"(gptr) : "memory");
#endif
}

__device__ __forceinline__ void wait_async_le2() {
#if __has_builtin(__builtin_amdgcn_s_wait_asynccnt)
  __builtin_amdgcn_s_wait_asynccnt(2);
#else
  asm volatile("s_wait_asynccnt 2" ::: "memory");
#endif
}

__device__ __forceinline__ void wait_async_le0() {
#if __has_builtin(__builtin_amdgcn_s_wait_asynccnt)
  __builtin_amdgcn_s_wait_asynccnt(0);
#else
  asm volatile("s_wait_asynccnt 0" ::: "memory");
#endif
}

// torch-style clamped BCE term: t*max(log p,-100) + (1-t)*max(log1p(-p),-100)
__device__ __forceinline__ float bce_term(float p, float t) {
  float lp = fmaxf(logf(p),      -100.0f);
  float l1 = fmaxf(log1pf(-p),   -100.0f);
  return l1 + t * (lp - l1);
}

// ---------------------------------------------------------------------------
// Kernel 1: per-block partial sums of F_c (target count) and S_c (log terms).
// Double-buffered async-to-LDS pipeline; each lane consumes exactly the 16 B
// it staged (conflict-free ds_load_b128, no cross-wave barriers needed since
// ASYNCcnt is per-wave and lanes never share staged data).
// ---------------------------------------------------------------------------
__global__ __launch_bounds__(ICF_BLOCK)
void icf_partial_kernel(const float* __restrict__ inp,
                        const float* __restrict__ tgt,
                        float* __restrict__ partial,   // [gridDim.x][16]
                        int nChunks) {
  __shared__ float4 bIn[2][ICF_BLOCK];   // 8 KB
  __shared__ float4 bTg[2][ICF_BLOCK];   // 8 KB
  __shared__ float  wred[8][16];         // per-wave partials

  const int tid  = threadIdx.x;
  const int lane = tid & 31;
  const int wave = tid >> 5;
  const int stride = gridDim.x;

  float accF[8], accS[8];
#pragma unroll
  for (int j = 0; j < 8; ++j) { accF[j] = 0.0f; accS[j] = 0.0f; }

  int c0  = blockIdx.x;
  int buf = 0;
  if (c0 < nChunks) {
    long long base = (long long)c0 * CHUNK_FLTS + tid * 4;
    async_ld16(inp + base, &bIn[0][tid]);
    async_ld16(tgt + base, &bTg[0][tid]);
  }

  for (int c = c0; c < nChunks; c += stride) {
    int nxt = c + stride;
    if (nxt < nChunks) {
      long long nb = (long long)nxt * CHUNK_FLTS + tid * 4;
      async_ld16(inp + nb, &bIn[buf ^ 1][tid]);
      async_ld16(tgt + nb, &bTg[buf ^ 1][tid]);
      wait_async_le2();      // current pair (issued last iter) has landed
    } else {
      wait_async_le0();
    }

    float4 p = bIn[buf][tid];   // ds_load_b128, reads own staged bytes
    float4 t = bTg[buf][tid];

    float s0 = bce_term(p.x, t.x);
    float s1 = bce_term(p.y, t.y);
    float s2 = bce_term(p.z, t.z);
    float s3 = bce_term(p.w, t.w);

    // chunk base is a multiple of 8 floats => lane parity fixes class group
    if ((tid & 1) == 0) {
      accS[0] += s0;  accS[1] += s1;  accS[2] += s2;  accS[3] += s3;
      accF[0] += t.x; accF[1] += t.y; accF[2] += t.z; accF[3] += t.w;
    } else {
      accS[4] += s0;  accS[5] += s1;  accS[6] += s2;  accS[7] += s3;
      accF[4] += t.x; accF[5] += t.y; accF[6] += t.z; accF[7] += t.w;
    }
    buf ^= 1;
  }

  // deterministic wave32 reduction (XOR shuffles), then cross-wave via LDS
#pragma unroll
  for (int j = 0; j < 8; ++j) {
    float f = accF[j], s = accS[j];
#pragma unroll
    for (int m = 16; m >= 1; m >>= 1) {
      f += __shfl_xor(f, m, 32);
      s += __shfl_xor(s, m, 32);
    }
    if (lane == 0) { wred[wave][j] = f; wred[wave][8 + j] = s; }
  }
  __syncthreads();

  if (tid < 16) {
    float v = 0.0f;
#pragma unroll
    for (int w = 0; w < 8; ++w) v += wred[w][tid];
    partial[(long long)blockIdx.x * 16 + tid] = v;   // [0..7]=F, [8..15]=S
  }
}

// ---------------------------------------------------------------------------
// Kernel 2: deterministic reduction of per-block partials (double precision),
// tail handling, and final scalar.
// ---------------------------------------------------------------------------
__global__ __launch_bounds__(ICF_BLOCK)
void icf_final_kernel(const float* __restrict__ partial,
                      const float* __restrict__ inp,
                      const float* __restrict__ tgt,
                      float* __restrict__ out,
                      long long total, int nChunks, int nBlocks) {
  __shared__ double red[ICF_BLOCK];
  const int tid = threadIdx.x;
  const int j = tid & 15;        // which of the 16 stats
  const int g = tid >> 4;        // 16 groups over blocks

  double a = 0.0;
  for (int b = g; b < nBlocks; b += 16)
    a += (double)partial[(long long)b * 16 + j];
  red[tid] = a;
  __syncthreads();

#pragma unroll
  for (int s = 8; s >= 1; s >>= 1) {
    if (g < s) red[tid] += red[tid + 16 * s];
    __syncthreads();
  }

  if (tid == 0) {
    double F[8], S[8];
    for (int k = 0; k < 8; ++k) { F[k] = red[k]; S[k] = red[8 + k]; }

    // scalar tail (only if total % 1024 != 0)
    long long tail = (long long)nChunks * CHUNK_FLTS;
    for (long long i = tail; i < total; ++i) {
      float p = inp[i], t = tgt[i];
      int cls = (int)(i & 7);
      float lp = fmaxf(logf(p),    -100.0f);
      float l1 = fmaxf(log1pf(-p), -100.0f);
      S[cls] += (double)(l1 + t * (lp - l1));
      F[cls] += (double)t;
    }

    double acc = 0.0;
    for (int k = 0; k < 8; ++k)
      acc -= S[k] / (F[k] + 0.001);          // weights = 1/(freq + EPSILON)
    out[0] = (float)(acc / (double)total);   // mean over B*C elements
  }
}

// ---------------------------------------------------------------------------
extern "C" void kernel_launch(void* const* d_in, const int* in_sizes, int n_in,
                              void* d_out, int out_size, void* d_ws, size_t ws_size,
                              hipStream_t stream) {
  const float* inp = (const float*)d_in[0];
  const float* tgt = (const float*)d_in[1];
  float* out = (float*)d_out;
  float* ws  = (float*)d_ws;

  long long total = (long long)in_sizes[0];          // B*C floats per array
  int nChunks = (int)(total / CHUNK_FLTS);           // full 1024-float chunks

  int nBlocks = ICF_GRID;
  size_t need = (size_t)nBlocks * 16 * sizeof(float);
  if (ws_size < need) {
    nBlocks = (int)(ws_size / (16 * sizeof(float)));
    if (nBlocks < 1) nBlocks = 1;
  }
  if (nChunks > 0 && nBlocks > nChunks) nBlocks = nChunks;
  if (nChunks == 0) nBlocks = 1;

  icf_partial_kernel<<<nBlocks, ICF_BLOCK, 0, stream>>>(inp, tgt, ws, nChunks);
  icf_final_kernel<<<1, ICF_BLOCK, 0, stream>>>(ws, inp, tgt, out,
                                                total, nChunks, nBlocks);
}